// StockGNN_30760555774221
// MI455X (gfx1250) — compile-verified
//
#include <hip/hip_runtime.h>

#define DH 128
#define NGRAPH 64

typedef __attribute__((ext_vector_type(2))) float v2f;
typedef __attribute__((ext_vector_type(8))) float v8f;

// ---------------- utility ----------------
__global__ void zero_f32(float* __restrict__ p, long n) {
    long i = (long)blockIdx.x * blockDim.x + threadIdx.x;
    if (i < n) p[i] = 0.0f;
}

// ---------------- degree / norm ----------------
__global__ void deg_edges(const int* __restrict__ dst, float* __restrict__ deg, int E) {
    int e = blockIdx.x * blockDim.x + threadIdx.x;
    if (e < E) atomicAdd(&deg[dst[e]], 1.0f);
}

__global__ void dinv_kernel(const float* __restrict__ deg, float* __restrict__ dinv, int n) {
    int i = blockIdx.x * blockDim.x + threadIdx.x;
    if (i < n) dinv[i] = rsqrtf(deg[i] + 1.0f);   // +1 = self loop; always > 0
}

// ---------------- WMMA GEMM: H[nrows,128] = X[nrows,128] @ W[128,128] ----------------
// block = 256 threads = 8 waves; wave w computes column tile [16w, 16w+16)
// W staged fully in LDS (64KB); X staged 16 rows at a time (8KB) via async-to-LDS
__global__ __launch_bounds__(256, 1)
void gcn_gemm_wmma(const float* __restrict__ X, const float* __restrict__ W,
                   float* __restrict__ H, int nrows) {
    __shared__ float sW[DH * DH];
    __shared__ float sX[16 * DH];
    const int tid = threadIdx.x;
    for (int i = tid; i < DH * DH; i += 256) sW[i] = W[i];

    const int wave = tid >> 5;
    const int lane = tid & 31;
    const int hlf  = lane >> 4;       // 0: lanes 0-15, 1: lanes 16-31
    const int l16  = lane & 15;
    const int col0 = wave * 16;
    const int tilesTotal = (nrows + 15) >> 4;

    for (int tile = blockIdx.x; tile < tilesTotal; tile += gridDim.x) {
        const int row0 = tile << 4;
        const bool fullTile = (row0 + 16 <= nrows);
        __syncthreads();              // protect sX readers from previous iter
        if (fullTile) {
            // async DMA the 16x128 X tile into LDS: 512 b128 packets, 2 per thread
            #pragma unroll
            for (int i = tid; i < 16 * DH / 4; i += 256) {
                int r = i >> 5, c4 = i & 31;            // 32 float4 per row
                unsigned ldsoff = (unsigned)(size_t)&sX[r * DH + c4 * 4];
                const float* g = X + (size_t)(row0 + r) * DH + c4 * 4;
                asm volatile("global_load_async_to_lds_b128 %0, %1, off"
                             :: "v"(ldsoff), "v"(g) : "memory");
            }
            asm volatile("s_wait_asynccnt 0x0" ::: "memory");
        } else {
            for (int i = tid; i < 16 * DH; i += 256) {
                int r = i >> 7, c = i & (DH - 1);
                int gr = row0 + r;
                sX[i] = (gr < nrows) ? X[(size_t)gr * DH + c] : 0.0f;
            }
        }
        __syncthreads();

        v8f acc = {0.f, 0.f, 0.f, 0.f, 0.f, 0.f, 0.f, 0.f};
        #pragma unroll 4
        for (int k = 0; k < DH; k += 4) {
            // A 16x4 fp32: lanes hold M=l16; VGPR0 = K (k+0 | k+2), VGPR1 = K (k+1 | k+3)
            v2f a, b;
            const int ka = k + hlf * 2;
            a.x = sX[l16 * DH + ka];
            a.y = sX[l16 * DH + ka + 1];
            // B 4x16 fp32: row K striped across lanes; VGPR0 = K (k+0 | k+2), VGPR1 = K (k+1 | k+3)
            b.x = sW[(size_t)(ka)     * DH + col0 + l16];
            b.y = sW[(size_t)(ka + 1) * DH + col0 + l16];
            acc = __builtin_amdgcn_wmma_f32_16x16x4_f32(
                false, a, false, b, (short)0, acc, false, false);
        }

        // C/D 16x16 f32 layout: VGPR v -> row (v + 8*hlf), col l16
        if (fullTile) {
            float* Hp = H + (size_t)(row0 + hlf * 8) * DH + col0 + l16;
            #pragma unroll
            for (int v = 0; v < 8; ++v) Hp[(size_t)v * DH] = acc[v];
        } else {
            const int rbase = row0 + hlf * 8;
            #pragma unroll
            for (int v = 0; v < 8; ++v) {
                int r = rbase + v;
                if (r < nrows) H[(size_t)r * DH + col0 + l16] = acc[v];
            }
        }
    }
}

// ---------------- self-loop init: agg[i] = dinv[i]^2 * hw[i] ----------------
__global__ void self_init(const float* __restrict__ hw, const float* __restrict__ dinv,
                          float* __restrict__ agg, int n) {
    long t = (long)blockIdx.x * blockDim.x + threadIdx.x;
    int node = (int)(t >> 5);
    if (node >= n) return;
    int ch = (int)(t & 31);
    float nrm = dinv[node] * dinv[node];
    float4 v = ((const float4*)(hw + (size_t)node * DH))[ch];
    v.x *= nrm; v.y *= nrm; v.z *= nrm; v.w *= nrm;
    ((float4*)(agg + (size_t)node * DH))[ch] = v;
}

// ---------------- edge scatter: one wave per edge, coalesced float4 gather ----------------
// hw (51MB) and agg (51MB) are L2-resident (192MB L2): gather + atomics hit L2.
__global__ void edge_scatter(const float* __restrict__ hw,
                             const int* __restrict__ src, const int* __restrict__ dst,
                             const float* __restrict__ dinv,
                             float* __restrict__ agg, int E) {
    long t = (long)blockIdx.x * blockDim.x + threadIdx.x;
    int e = (int)(t >> 5);
    if (e >= E) return;
    int ch = (int)(t & 31);
    int s = src[e], d = dst[e];
    float nrm = dinv[s] * dinv[d];
    float4 v = ((const float4*)(hw + (size_t)s * DH))[ch];
    float* out = agg + (size_t)d * DH + ch * 4;
    atomicAdd(out + 0, v.x * nrm);
    atomicAdd(out + 1, v.y * nrm);
    atomicAdd(out + 2, v.z * nrm);
    atomicAdd(out + 3, v.w * nrm);
}

// ---------------- batchnorm stats: per-feature sum & sumsq ----------------
__global__ void bn_stats(const float* __restrict__ h, float* __restrict__ stats, int n) {
    int f = threadIdx.x;  // 128 threads
    float s = 0.0f, ss = 0.0f;
    for (int i = blockIdx.x; i < n; i += gridDim.x) {
        float v = h[(size_t)i * DH + f];
        s += v; ss += v * v;
    }
    atomicAdd(&stats[f], s);
    atomicAdd(&stats[DH + f], ss);
}

__global__ void bn_apply_relu(float* __restrict__ h, const float* __restrict__ stats,
                              const float* __restrict__ gamma, const float* __restrict__ beta,
                              int n) {
    long t = (long)blockIdx.x * blockDim.x + threadIdx.x;
    if (t >= (long)n * DH) return;
    int f = (int)(t & (DH - 1));
    float invn = 1.0f / (float)n;
    float mean = stats[f] * invn;
    float var  = stats[DH + f] * invn - mean * mean;
    float istd = rsqrtf(var + 1e-5f);
    float v = (h[t] - mean) * istd * gamma[f] + beta[f];
    h[t] = v > 0.0f ? v : 0.0f;
}

// ---------------- global mean pool ----------------
__global__ void pool_kernel(const float* __restrict__ h, const int* __restrict__ batch,
                            float* __restrict__ pooled, float* __restrict__ cnt, int n) {
    long t = (long)blockIdx.x * blockDim.x + threadIdx.x;
    int node = (int)(t >> 5);
    if (node >= n) return;
    int ch = (int)(t & 31);
    int g = batch[node];
    float4 v = ((const float4*)(h + (size_t)node * DH))[ch];
    float* p = pooled + (size_t)g * DH + ch * 4;
    atomicAdd(p + 0, v.x);
    atomicAdd(p + 1, v.y);
    atomicAdd(p + 2, v.z);
    atomicAdd(p + 3, v.w);
    if (ch == 0) atomicAdd(&cnt[g], 1.0f);
}

// ---------------- final MLP: out[g] = relu(pooled_mean @ w1 + b1) @ w2 + b2 ----------------
__global__ void mlp_kernel(const float* __restrict__ pooled, const float* __restrict__ cnt,
                           const float* __restrict__ w1, const float* __restrict__ b1,
                           const float* __restrict__ w2, const float* __restrict__ b2,
                           float* __restrict__ out) {
    __shared__ float red[DH];
    int g = blockIdx.x, j = threadIdx.x;  // 128 threads
    float invc = 1.0f / fmaxf(cnt[g], 1.0f);
    float acc = b1[j];
    #pragma unroll 8
    for (int k = 0; k < DH; ++k)
        acc += (pooled[(size_t)g * DH + k] * invc) * w1[(size_t)k * DH + j];
    red[j] = fmaxf(acc, 0.0f) * w2[j];
    __syncthreads();
    for (int s = DH / 2; s > 0; s >>= 1) {
        if (j < s) red[j] += red[j + s];
        __syncthreads();
    }
    if (j == 0) out[g] = red[0] + b2[0];
}

extern "C" void kernel_launch(void* const* d_in, const int* in_sizes, int n_in,
                              void* d_out, int out_size, void* d_ws, size_t ws_size,
                              hipStream_t stream) {
    const float* x       = (const float*)d_in[0];
    const int*   eidx    = (const int*)d_in[1];
    const int*   batch   = (const int*)d_in[2];
    const float* conv_w  = (const float*)d_in[3];
    const float* bn_g    = (const float*)d_in[5];
    const float* bn_b    = (const float*)d_in[6];
    const float* w1      = (const float*)d_in[7];
    const float* b1      = (const float*)d_in[8];
    const float* w2      = (const float*)d_in[9];
    const float* b2      = (const float*)d_in[10];
    float* out = (float*)d_out;

    const int N = in_sizes[0] / DH;       // 100000
    const int E = in_sizes[1] / 2;        // 3200000
    const int* src = eidx;
    const int* dst = eidx + E;

    float* ws      = (float*)d_ws;
    float* bufA    = ws;                              // N*DH  (GEMM output hw)
    float* bufB    = bufA + (size_t)N * DH;           // N*DH  (agg / layer activations)
    float* deg     = bufB + (size_t)N * DH;           // N
    float* dinv    = deg + N;                         // N
    float* stats   = dinv + N;                        // 2*DH
    float* pooled  = stats + 2 * DH;                  // NGRAPH*DH
    float* cnt     = pooled + (size_t)NGRAPH * DH;    // NGRAPH

    // degree / symmetric norm
    zero_f32<<<(N + 255) / 256, 256, 0, stream>>>(deg, N);
    deg_edges<<<(E + 255) / 256, 256, 0, stream>>>(dst, deg, E);
    dinv_kernel<<<(N + 255) / 256, 256, 0, stream>>>(deg, dinv, N);

    const long nodeVec = (long)N * 32;   // 32 float4 lanes per node

    for (int layer = 0; layer < 3; ++layer) {
        const float* Xin = (layer == 0) ? x : bufB;
        const float* W   = conv_w + (size_t)layer * DH * DH;

        int tiles = (N + 15) / 16;
        int gblocks = tiles < 2048 ? tiles : 2048;
        gcn_gemm_wmma<<<gblocks, 256, 0, stream>>>(Xin, W, bufA, N);

        self_init<<<(int)((nodeVec + 255) / 256), 256, 0, stream>>>(bufA, dinv, bufB, N);
        edge_scatter<<<(int)(((long)E * 32 + 255) / 256), 256, 0, stream>>>(
            bufA, src, dst, dinv, bufB, E);

        zero_f32<<<1, 256, 0, stream>>>(stats, 2 * DH);
        bn_stats<<<512, DH, 0, stream>>>(bufB, stats, N);
        bn_apply_relu<<<(int)(((long)N * DH + 255) / 256), 256, 0, stream>>>(
            bufB, stats, bn_g + layer * DH, bn_b + layer * DH, N);
    }

    zero_f32<<<(NGRAPH * DH + NGRAPH + 255) / 256, 256, 0, stream>>>(pooled, NGRAPH * DH + NGRAPH);
    pool_kernel<<<(int)((nodeVec + 255) / 256), 256, 0, stream>>>(bufB, batch, pooled, cnt, N);
    mlp_kernel<<<NGRAPH, DH, 0, stream>>>(pooled, cnt, w1, b1, w2, b2, out);
}